// YoloLoss_85177791414999
// MI455X (gfx1250) — compile-verified
//
#include <hip/hip_runtime.h>
#include <hip/hip_bf16.h>

// YOLO loss, MI455X (gfx1250, wave32).
// Memory-bound streaming reduction (~174 MB -> 5 floats, roofline ~7.5us @ 23.3 TB/s).
// Kernel 1: async-tensor path (global_load_async_to_lds_b128 + s_wait_asynccnt) stages
//           512-cell pred tiles into LDS; per-cell loss math; wave32 shfl reduce;
//           one float4 partial per block -> d_ws.
// Kernel 2: single wave32, V_WMMA_F32_16X16X4_F32 accumulator-chained row-sum reduction
//           of the 1568 float4 partials; lane 0 writes the 5 outputs.

#define CELLS   802816      // 4096 * 14 * 14
#define TILE    512
#define NBLK    1568        // CELLS / TILE (exact)
#define STEPS   392         // NBLK / 4    (exact)

typedef __attribute__((ext_vector_type(2))) float v2f;
typedef __attribute__((ext_vector_type(8))) float v8f;

// ---- gfx1250 async global->LDS copy (ASYNCcnt-tracked, CDNA5 ISA & /08) ----
// Single __shared__ array => group segment offset 0, so the LDS byte offset we
// pass in VDST equals the array byte offset.
__device__ __forceinline__ void async_g2l_b128(const float* gsrc, unsigned lds_byte_off) {
  asm volatile("global_load_async_to_lds_b128 %0, %1, off"
               :: "v"(lds_byte_off), "v"(gsrc)
               : "memory");
}
__device__ __forceinline__ void wait_async0() {
  asm volatile("s_wait_asynccnt 0x0" ::: "memory");
}

__device__ __forceinline__ float iou_xyxy(float ax0, float ay0, float ax1, float ay1,
                                          float bx0, float by0, float bx1, float by1) {
  float ltx = fmaxf(ax0, bx0), lty = fmaxf(ay0, by0);
  float rbx = fminf(ax1, bx1), rby = fminf(ay1, by1);
  float w = fmaxf(rbx - ltx, 0.0f), h = fmaxf(rby - lty, 0.0f);
  float inter = w * h;
  float areaA = (ax1 - ax0) * (ay1 - ay0);
  float areaB = (bx1 - bx0) * (by1 - by0);
  return inter / (areaA + areaB - inter);
}

__global__ __launch_bounds__(256)
void yolo_partial(const float* __restrict__ pred,
                  const float4* __restrict__ tbox,
                  const float4* __restrict__ tcls4,
                  const unsigned char* __restrict__ objmap,
                  float4* __restrict__ part) {
  __shared__ float lds_pred[TILE * 30];            // 61440 B, offset 0 in group segment
  const int t     = threadIdx.x;
  const int tile  = blockIdx.x;
  const int cell0 = tile * TILE;
  const float* src = pred + (size_t)cell0 * 30;

  // Stage pred tile: 3840 x b128 chunks, 15 per thread, fully coalesced.
#pragma unroll
  for (int j = 0; j < 15; ++j) {
    const int q = j * 256 + t;                     // 16B chunk id, 0..3839
    async_g2l_b128(src + q * 4, (unsigned)(q * 16));
  }
  wait_async0();
  __syncthreads();

  float rs = 0.0f, os = 0.0f, ns = 0.0f, cs = 0.0f;
#pragma unroll
  for (int half = 0; half < 2; ++half) {
    const int c    = t + half * 256;               // cell within tile
    const int cell = cell0 + c;
    const float* pr = &lds_pred[c * 30];           // 30-dword stride: conflict-free mod 64 banks

    const float m = objmap[cell] ? 1.0f : 0.0f;
    const float4 tb = tbox[cell];

    const float b1x = pr[0], b1y = pr[1], b1w = pr[2], b1h = pr[3], b1c = pr[4];
    const float b2x = pr[5], b2y = pr[6], b2w = pr[7], b2h = pr[8], b2c = pr[9];

    // class loss
    float cls = 0.0f;
#pragma unroll
    for (int k = 0; k < 5; ++k) {
      const float4 t4 = tcls4[cell * 5 + k];
      const float d0 = pr[10 + 4 * k + 0] - t4.x;
      const float d1 = pr[10 + 4 * k + 1] - t4.y;
      const float d2 = pr[10 + 4 * k + 2] - t4.z;
      const float d3 = pr[10 + 4 * k + 3] - t4.w;
      cls += d0 * d0 + d1 * d1 + d2 * d2 + d3 * d3;
    }
    cs += m * cls;

    // no-object loss: ((1-m)*conf)^2 == (1-m)*conf^2 for m in {0,1}
    ns += (1.0f - m) * (b1c * b1c + b2c * b2c);

    // xywh -> xyxy (xy scaled by 1/S_GRID), IoU vs target
    const float inv = 1.0f / 14.0f;
    float cx, cy;
    cx = b1x * inv; cy = b1y * inv;
    const float ax0 = cx - 0.5f * b1w, ay0 = cy - 0.5f * b1h;
    const float ax1 = cx + 0.5f * b1w, ay1 = cy + 0.5f * b1h;
    cx = b2x * inv; cy = b2y * inv;
    const float bx0 = cx - 0.5f * b2w, by0 = cy - 0.5f * b2h;
    const float bx1 = cx + 0.5f * b2w, by1 = cy + 0.5f * b2h;
    cx = tb.x * inv; cy = tb.y * inv;
    const float tx0 = cx - 0.5f * tb.z, ty0 = cy - 0.5f * tb.w;
    const float tx1 = cx + 0.5f * tb.z, ty1 = cy + 0.5f * tb.w;

    const float iou1 = iou_xyxy(ax0, ay0, ax1, ay1, tx0, ty0, tx1, ty1);
    const float iou2 = iou_xyxy(bx0, by0, bx1, by1, tx0, ty0, tx1, ty1);
    const bool sel = iou1 >= iou2;

    const float px = sel ? b1x : b2x, py = sel ? b1y : b2y;
    const float pw = sel ? b1w : b2w, ph = sel ? b1h : b2h;
    const float pc = sel ? b1c : b2c;
    const float biou = sel ? iou1 : iou2;

    const float dx = px - tb.x, dy = py - tb.y;
    const float dw = sqrtf(pw) - sqrtf(tb.z);
    const float dh = sqrtf(ph) - sqrtf(tb.w);
    rs += m * (dx * dx + dy * dy + dw * dw + dh * dh);

    const float dc = pc - biou;
    os += m * dc * dc;
  }

  // wave32 butterfly reduction
#pragma unroll
  for (int off = 16; off > 0; off >>= 1) {
    rs += __shfl_xor(rs, off, 32);
    os += __shfl_xor(os, off, 32);
    ns += __shfl_xor(ns, off, 32);
    cs += __shfl_xor(cs, off, 32);
  }

  __syncthreads();                                 // all waves done reading the tile buffer
  const int lane = t & 31, wv = t >> 5;
  if (lane == 0) {
    lds_pred[wv * 4 + 0] = rs; lds_pred[wv * 4 + 1] = os;
    lds_pred[wv * 4 + 2] = ns; lds_pred[wv * 4 + 3] = cs;
  }
  __syncthreads();
  if (t == 0) {
    float a = 0.0f, b = 0.0f, c2 = 0.0f, d = 0.0f;
#pragma unroll
    for (int w = 0; w < 8; ++w) {
      a  += lds_pred[w * 4 + 0];
      b  += lds_pred[w * 4 + 1];
      c2 += lds_pred[w * 4 + 2];
      d  += lds_pred[w * 4 + 3];
    }
    part[tile] = make_float4(a, b, c2, d);
  }
}

// Final reduction on the CDNA5 matrix pipe: D = A(16x4) x Ones(4x16) + D sums
// rows of A into D. Rows 0..3 of A carry {reg,obj,noobj,cls}; per the ISA A
// layout, lanes 0-15 supply K=0,1 and lanes 16-31 supply K=2,3. After 392
// accumulation steps, lane 0 holds the component sums in D VGPRs 0..3.
__global__ __launch_bounds__(32)
void yolo_final(const float* __restrict__ P /* NBLK*4 floats */, float* __restrict__ out) {
  const int lane  = threadIdx.x;
  const int r     = lane & 15;
  const int hi    = (lane >> 4) & 1;
  const bool valid = (r < 4);
  const int comp  = valid ? r : 0;

  v8f acc = {0.f, 0.f, 0.f, 0.f, 0.f, 0.f, 0.f, 0.f};
  v2f ones; ones[0] = 1.0f; ones[1] = 1.0f;

  for (int s = 0; s < STEPS; ++s) {
    const int blk = 4 * s + 2 * hi;                // float4-partial index for K=0/2
    const float a0 = P[blk * 4 + comp];            // A[r, K]   (K = 0 or 2)
    const float a1 = P[(blk + 1) * 4 + comp];      // A[r, K+1]
    v2f A;
    A[0] = valid ? a0 : 0.0f;                      // cndmask, EXEC stays all-1s
    A[1] = valid ? a1 : 0.0f;
    acc = __builtin_amdgcn_wmma_f32_16x16x4_f32(false, A, false, ones,
                                                (short)0, acc, false, false);
  }

  if (lane == 0) {
    const float reg = acc[0], obj = acc[1], noobj = acc[2], cls = acc[3];
    const float total = cls + 0.5f * noobj + 5.0f * reg + obj;
    const float invN = 1.0f / 4096.0f;
    out[0] = total * invN;
    out[1] = reg   * invN;
    out[2] = obj   * invN;
    out[3] = noobj * invN;
    out[4] = cls   * invN;
  }
}

extern "C" void kernel_launch(void* const* d_in, const int* in_sizes, int n_in,
                              void* d_out, int out_size, void* d_ws, size_t ws_size,
                              hipStream_t stream) {
  (void)in_sizes; (void)n_in; (void)out_size; (void)ws_size;
  const float* pred         = (const float*)d_in[0];
  const float4* tbox        = (const float4*)d_in[1];
  const float4* tcls4       = (const float4*)d_in[2];
  const unsigned char* objm = (const unsigned char*)d_in[3];  // jax bool = 1 byte
  float4* part = (float4*)d_ws;                                // NBLK*16 B = 25 KB
  float*  out  = (float*)d_out;

  yolo_partial<<<NBLK, 256, 0, stream>>>(pred, tbox, tcls4, objm, part);
  yolo_final<<<1, 32, 0, stream>>>((const float*)d_ws, out);
}